// simple_GC_DEC_PyG_58703613002030
// MI455X (gfx1250) — compile-verified
//
#include <hip/hip_runtime.h>
#include <math.h>

#define N_NODES   100000
#define N_EDGES   1600000
#define NFEAT     128
#define NHID      64
#define N_CLUST   20
#define ALPHA_F   0.2f

typedef __attribute__((ext_vector_type(2))) float v2f;
typedef __attribute__((ext_vector_type(8))) float v8f;

// ---------------------------------------------------------------------------
// K0: deg = 1.0 (self-loop weight), out_z = bias broadcast
// ---------------------------------------------------------------------------
__global__ void k_init(float* __restrict__ deg, float* __restrict__ out_z,
                       const float* __restrict__ b) {
    int tid = blockIdx.x * blockDim.x + threadIdx.x;
    if (tid >= N_NODES * NHID) return;
    int f = tid & (NHID - 1);
    out_z[tid] = b[f];
    if (f == 0) deg[tid >> 6] = 1.0f;
}

// ---------------------------------------------------------------------------
// K1: support = x @ W via V_WMMA_F32_16X16X4_F32 (fp32 in, fp32 acc)
// One wave -> 16 rows x 64 cols (4 accumulators share one A fragment).
// ---------------------------------------------------------------------------
__global__ void k_gemm(const float* __restrict__ x, const float* __restrict__ W,
                       float* __restrict__ support) {
    int wave = (int)((blockIdx.x * (unsigned)blockDim.x + threadIdx.x) >> 5);
    int lane = threadIdx.x & 31;
    if (wave * 16 >= N_NODES) return;          // uniform per wave
    int half = lane >> 4;
    int m    = lane & 15;
    const float* xr = x + (size_t)(wave * 16 + m) * NFEAT;

    v8f acc0 = {}, acc1 = {}, acc2 = {}, acc3 = {};
    for (int k0 = 0; k0 < NFEAT; k0 += 4) {
        int ka = k0 + 2 * half;                // A: VGPR v holds K = 2*half + v
        v2f a;  a.x = xr[ka];      a.y = xr[ka + 1];
        const float* wp = W + (size_t)ka * NHID + m;   // B: N = m, K = 2*half + v
        v2f b0; b0.x = wp[0];      b0.y = wp[NHID + 0];
        v2f b1; b1.x = wp[16];     b1.y = wp[NHID + 16];
        v2f b2; b2.x = wp[32];     b2.y = wp[NHID + 32];
        v2f b3; b3.x = wp[48];     b3.y = wp[NHID + 48];
        acc0 = __builtin_amdgcn_wmma_f32_16x16x4_f32(false, a, false, b0, (short)0, acc0, false, false);
        acc1 = __builtin_amdgcn_wmma_f32_16x16x4_f32(false, a, false, b1, (short)0, acc1, false, false);
        acc2 = __builtin_amdgcn_wmma_f32_16x16x4_f32(false, a, false, b2, (short)0, acc2, false, false);
        acc3 = __builtin_amdgcn_wmma_f32_16x16x4_f32(false, a, false, b3, (short)0, acc3, false, false);
    }
    // D layout: VGPR r, lane (half,m) -> element (M = r + 8*half, N = m)
    int rbase = wave * 16 + 8 * half;
#pragma unroll
    for (int r = 0; r < 8; ++r) {
        float* o = support + (size_t)(rbase + r) * NHID + m;
        o[0]  = acc0[r];
        o[16] = acc1[r];
        o[32] = acc2[r];
        o[48] = acc3[r];
    }
}

// ---------------------------------------------------------------------------
// K2: weighted in-degree scatter
// ---------------------------------------------------------------------------
__global__ void k_deg(const int* __restrict__ ei, const float* __restrict__ ew,
                      float* __restrict__ deg) {
    int e = blockIdx.x * blockDim.x + threadIdx.x;
    if (e >= N_EDGES) return;
    atomicAdd(&deg[ei[N_EDGES + e]], ew[e]);
}

// ---------------------------------------------------------------------------
// K3: dinv = rsqrt(deg) (deg > 0 guaranteed by self-loop, keep ref guard)
// ---------------------------------------------------------------------------
__global__ void k_dinv(const float* __restrict__ deg, float* __restrict__ dinv) {
    int i = blockIdx.x * blockDim.x + threadIdx.x;
    if (i >= N_NODES) return;
    float d = deg[i];
    dinv[i] = (d > 0.0f) ? rsqrtf(d) : 0.0f;
}

// ---------------------------------------------------------------------------
// K4: message scatter-add. 16 lanes per edge, float4 gather + 4 atomics.
// support (25.6MB) and out_z (25.6MB) are L2-resident on MI455X (192MB L2).
// ---------------------------------------------------------------------------
__global__ void k_scatter(const int* __restrict__ ei, const float* __restrict__ ew,
                          const float* __restrict__ dinv,
                          const float* __restrict__ support,
                          float* __restrict__ out_z) {
    int tid = blockIdx.x * blockDim.x + threadIdx.x;
    int e = tid >> 4;
    if (e >= N_EDGES) return;
    int g   = (tid & 15) * 4;
    int src = ei[e];
    int dst = ei[N_EDGES + e];
    float norm = dinv[src] * ew[e] * dinv[dst];
    const float4 s = *(const float4*)(support + (size_t)src * NHID + g);
    float* o = out_z + (size_t)dst * NHID + g;
    atomicAdd(o + 0, norm * s.x);
    atomicAdd(o + 1, norm * s.y);
    atomicAdd(o + 2, norm * s.z);
    atomicAdd(o + 3, norm * s.w);
}

// ---------------------------------------------------------------------------
// K5: self-loop term, race-free (unique per node), after scatter completes
// ---------------------------------------------------------------------------
__global__ void k_selfloop(const float* __restrict__ dinv,
                           const float* __restrict__ support,
                           float* __restrict__ out_z) {
    int tid = blockIdx.x * blockDim.x + threadIdx.x;
    if (tid >= N_NODES * NHID) return;
    float di = dinv[tid >> 6];
    out_z[tid] += di * di * support[tid];
}

// ---------------------------------------------------------------------------
// K6: DEC Student-t head. One wave32 per node; butterfly shfl reductions.
// ---------------------------------------------------------------------------
__global__ void k_dec(const float* __restrict__ z, const float* __restrict__ mu,
                      float* __restrict__ q) {
    int wave = (int)((blockIdx.x * (unsigned)blockDim.x + threadIdx.x) >> 5);
    int lane = threadIdx.x & 31;
    if (wave >= N_NODES) return;               // uniform per wave
    const float* zr = z + (size_t)wave * NHID;
    float z0 = zr[lane];
    float z1 = zr[lane + 32];
    float sum = 0.0f, mine = 0.0f;
    for (int k = 0; k < N_CLUST; ++k) {
        float d0 = z0 - mu[k * NHID + lane];
        float d1 = z1 - mu[k * NHID + lane + 32];
        float d  = d0 * d0 + d1 * d1;
#pragma unroll
        for (int off = 16; off > 0; off >>= 1)
            d += __shfl_xor(d, off, 32);
        float base = 1.0f / (1.0f + d * (1.0f / ALPHA_F) + 1e-8f);
        float qv = 0.5f * powf(base, 1.0f + ALPHA_F);
        if (lane == k) mine = qv;
        sum += qv;                              // identical across lanes
    }
    if (lane < N_CLUST) q[(size_t)wave * N_CLUST + lane] = mine / sum;
}

// ---------------------------------------------------------------------------
extern "C" void kernel_launch(void* const* d_in, const int* in_sizes, int n_in,
                              void* d_out, int out_size, void* d_ws, size_t ws_size,
                              hipStream_t stream) {
    const float* x  = (const float*)d_in[0];
    const int*   ei = (const int*)  d_in[1];
    const float* ew = (const float*)d_in[2];
    const float* W  = (const float*)d_in[3];
    const float* b  = (const float*)d_in[4];
    const float* mu = (const float*)d_in[5];

    float* out_z = (float*)d_out;                              // [N, 64]
    float* out_q = out_z + (size_t)N_NODES * NHID;             // [N, 20]

    char*  ws      = (char*)d_ws;
    float* support = (float*)ws;                                          // N*64
    float* deg     = (float*)(ws + (size_t)N_NODES * NHID * sizeof(float));
    float* dinv    = deg + N_NODES;

    const int B = 256;
    k_init    <<<(N_NODES * NHID + B - 1) / B, B, 0, stream>>>(deg, out_z, b);
    k_gemm    <<<((N_NODES / 16) * 32 + B - 1) / B, B, 0, stream>>>(x, W, support);
    k_deg     <<<(N_EDGES + B - 1) / B, B, 0, stream>>>(ei, ew, deg);
    k_dinv    <<<(N_NODES + B - 1) / B, B, 0, stream>>>(deg, dinv);
    k_scatter <<<(N_EDGES * 16 + B - 1) / B, B, 0, stream>>>(ei, ew, dinv, support, out_z);
    k_selfloop<<<(N_NODES * NHID + B - 1) / B, B, 0, stream>>>(dinv, support, out_z);
    k_dec     <<<(N_NODES * 32 + B - 1) / B, B, 0, stream>>>(out_z, mu, out_q);
}